// attn_head_21045339750752
// MI455X (gfx1250) — compile-verified
//
#include <hip/hip_runtime.h>

typedef float v2f __attribute__((ext_vector_type(2)));
typedef float v8f __attribute__((ext_vector_type(8)));

#define GAT_N   8192
#define GAT_H   64
#define GAT_FIN 128

#if defined(__HIP_DEVICE_COMPILE__)
#if !__has_builtin(__builtin_amdgcn_wmma_f32_16x16x4_f32)
#error "wmma_f32_16x16x4_f32 builtin not available on this toolchain"
#endif
#endif

// ---- CDNA5 async global->LDS DMA (ASYNCcnt-tracked), per CDNA5 ISA
__device__ __forceinline__ void async_ld_b128(const float* g, float* l) {
  unsigned int  lofs = (unsigned int)(uintptr_t)l;   // generic LDS ptr: addr[31:0] = LDS offset
  unsigned long long ga = (unsigned long long)(uintptr_t)g;
  asm volatile("global_load_async_to_lds_b128 %0, %1, off" :: "v"(lofs), "v"(ga) : "memory");
}
__device__ __forceinline__ void async_ld_b32(const float* g, float* l) {
  unsigned int  lofs = (unsigned int)(uintptr_t)l;
  unsigned long long ga = (unsigned long long)(uintptr_t)g;
  asm volatile("global_load_async_to_lds_b32 %0, %1, off" :: "v"(lofs), "v"(ga) : "memory");
}
template <int N> __device__ __forceinline__ void s_wait_async() {
  asm volatile("s_wait_asynccnt %0" :: "i"(N) : "memory");
}

// -------------------- Kernel 1: seq_fts = seq @ W (f32 WMMA) --------------------
#define K1_ROWS      32
#define K1_SEQ_PITCH 132   // 128 + 4 pad -> stride%64==4, conflict-free A reads
#define K1_W_PITCH   72    // 64 + 8 pad  -> halves hit disjoint bank groups

__global__ __launch_bounds__(64) void gat_proj(const float* __restrict__ seq,
                                               const float* __restrict__ W,
                                               float* __restrict__ sf) {
  __shared__ float seqL[K1_ROWS * K1_SEQ_PITCH];
  __shared__ float WL[GAT_FIN * K1_W_PITCH];
  const int tid  = threadIdx.x;
  const int lane = tid & 31;
  const int wv   = tid >> 5;           // 2 waves, 16 rows each
  const int Mloc = lane & 15;
  const int hl   = lane >> 4;
  const int rowBlock = blockIdx.x * K1_ROWS;

  // async-stage W (128x64, pitch 72) and seq tile (32x128, pitch 132)
  for (int idx = tid; idx < GAT_FIN * (GAT_H / 4); idx += 64) {
    int r = idx >> 4, q = idx & 15;
    async_ld_b128(W + (size_t)r * GAT_H + q * 4, WL + r * K1_W_PITCH + q * 4);
  }
  for (int idx = tid; idx < K1_ROWS * (GAT_FIN / 4); idx += 64) {
    int r = idx >> 5, q = idx & 31;
    async_ld_b128(seq + (size_t)(rowBlock + r) * GAT_FIN + q * 4,
                  seqL + r * K1_SEQ_PITCH + q * 4);
  }
  s_wait_async<0>();
  __syncthreads();

  v8f c[4];
#pragma unroll
  for (int nt = 0; nt < 4; ++nt) c[nt] = {0.f, 0.f, 0.f, 0.f, 0.f, 0.f, 0.f, 0.f};

  const int arow = wv * 16 + Mloc;
  for (int kk = 0; kk < GAT_FIN / 4; ++kk) {
    const int cb = kk * 4 + 2 * hl;    // lane's K pair within the K=4 step
    v2f a;
    a.x = seqL[arow * K1_SEQ_PITCH + cb];
    a.y = seqL[arow * K1_SEQ_PITCH + cb + 1];
#pragma unroll
    for (int nt = 0; nt < 4; ++nt) {
      v2f b;
      b.x = WL[(cb)     * K1_W_PITCH + nt * 16 + Mloc];
      b.y = WL[(cb + 1) * K1_W_PITCH + nt * 16 + Mloc];
      c[nt] = __builtin_amdgcn_wmma_f32_16x16x4_f32(false, a, false, b,
                                                    (short)0, c[nt], false, false);
    }
  }
  // C tile: vgpr r holds (M = r + 8*half, N = lane&15)
#pragma unroll
  for (int r = 0; r < 8; ++r) {
    int grow = rowBlock + wv * 16 + r + 8 * hl;
#pragma unroll
    for (int nt = 0; nt < 4; ++nt)
      sf[(size_t)grow * GAT_H + nt * 16 + Mloc] = c[nt][r];
  }
}

// -------------------- Kernel 1b: f1/f2 per-row dots --------------------
__global__ __launch_bounds__(256) void gat_f1f2(const float* __restrict__ sf,
                                                const float* __restrict__ a1w,
                                                const float* __restrict__ a1b,
                                                const float* __restrict__ a2w,
                                                const float* __restrict__ a2b,
                                                float* __restrict__ f1,
                                                float* __restrict__ f2) {
  const int lane = threadIdx.x & 31;
  const int wv   = threadIdx.x >> 5;
  const int row  = blockIdx.x * 8 + wv;
  const float2 v  = *(const float2*)(sf + (size_t)row * GAT_H + lane * 2);
  const float2 w1 = *(const float2*)(a1w + lane * 2);
  const float2 w2 = *(const float2*)(a2w + lane * 2);
  float p1 = v.x * w1.x + v.y * w1.y;
  float p2 = v.x * w2.x + v.y * w2.y;
#pragma unroll
  for (int off = 16; off > 0; off >>= 1) {
    p1 += __shfl_down(p1, off, 32);
    p2 += __shfl_down(p2, off, 32);
  }
  if (lane == 0) {
    f1[row] = p1 + a1b[0];
    f2[row] = p2 + a2b[0];
  }
}

// -------------------- Kernel 2: streaming softmax-aggregation --------------------
// Triple-buffered async-to-LDS pipeline: chunk n+1 DMAs while chunk n computes.
#define K2_ROWS   64
#define KC        32                     // columns per chunk
#define ADJ_PITCH 36                     // 32+4: 36*M mod 64 -> 16 distinct banks (==0 mod 4)
#define SF_PITCH  72                     // 64+8: K halves land in disjoint bank groups
#define ADJ_F     (K2_ROWS * ADJ_PITCH)  // 2304 floats
#define SF_F      (KC * SF_PITCH)        // 2304 floats
#define F2_F      KC                     // 32 floats
#define BUF_F     (ADJ_F + SF_F + F2_F)  // 4640 floats per buffer
#define NCHUNK    (GAT_N / KC)           // 256
#define ISSUED    9                      // async ops per wave per chunk (4+4+1)

__device__ __forceinline__ void issue_chunk(const float* __restrict__ adj,
                                            const float* __restrict__ sf,
                                            const float* __restrict__ f2,
                                            int rowBlock, int j0, float* buf, int tid) {
  float* adjB = buf;
  float* sfB  = buf + ADJ_F;
  float* f2B  = buf + ADJ_F + SF_F;
#pragma unroll
  for (int i = 0; i < 4; ++i) {
    int idx = i * 128 + tid;
    { // adj tile: 64 rows x 8 quads (32 cols)
      int r = idx >> 3, q = idx & 7;
      async_ld_b128(adj + (size_t)(rowBlock + r) * GAT_N + j0 + q * 4,
                    adjB + r * ADJ_PITCH + q * 4);
    }
    { // sf chunk: 32 rows x 16 quads (64 cols)
      int r = idx >> 4, q = idx & 15;
      async_ld_b128(sf + (size_t)(j0 + r) * GAT_H + q * 4,
                    sfB + r * SF_PITCH + q * 4);
    }
  }
  // f2 chunk (32 floats); all lanes issue -> uniform per-wave ASYNCcnt
  // (duplicate writes carry identical values -> benign)
  async_ld_b32(f2 + j0 + (tid & 31), f2B + (tid & 31));
}

__global__ __launch_bounds__(128) void gat_agg(const float* __restrict__ adj,
                                               const float* __restrict__ sf,
                                               const float* __restrict__ f1,
                                               const float* __restrict__ f2,
                                               const float* __restrict__ bz,
                                               float* __restrict__ out) {
  extern __shared__ float smem[];        // 3 * BUF_F floats

  const int tid  = threadIdx.x;
  const int lane = tid & 31;
  const int wv   = tid >> 5;             // 4 waves x 16 rows
  const int Mloc = lane & 15;
  const int hl   = lane >> 4;
  const int rowBlock = blockIdx.x * K2_ROWS;
  const int arow = wv * 16 + Mloc;       // block-local A row
  const float f1r = f1[rowBlock + arow];

  v8f c[4];
#pragma unroll
  for (int nt = 0; nt < 4; ++nt) c[nt] = {0.f, 0.f, 0.f, 0.f, 0.f, 0.f, 0.f, 0.f};
  float s = 0.f;

  issue_chunk(adj, sf, f2, rowBlock, 0, smem, tid);

  for (int ci = 0; ci < NCHUNK; ++ci) {
    if (ci + 1 < NCHUNK) {
      issue_chunk(adj, sf, f2, rowBlock, (ci + 1) * KC,
                  smem + ((ci + 1) % 3) * BUF_F, tid);
      s_wait_async<ISSUED>();            // current chunk landed; next still in flight
    } else {
      s_wait_async<0>();
    }
    __syncthreads();                     // all waves' DMAs for this chunk visible

    const float* buf  = smem + (ci % 3) * BUF_F;
    const float* adjB = buf;
    const float* sfB  = buf + ADJ_F;
    const float* f2B  = buf + ADJ_F + SF_F;

    for (int kk = 0; kk < KC / 4; ++kk) {
      const int cb = kk * 4 + 2 * hl;
      // A = adj * exp(leaky_relu(f1+f2)); masked entries -> exact 0
      float a0 = adjB[arow * ADJ_PITCH + cb];
      float a1 = adjB[arow * ADJ_PITCH + cb + 1];
      float x0 = f1r + f2B[cb];
      float x1 = f1r + f2B[cb + 1];
      x0 = (x0 > 0.f) ? x0 : 0.2f * x0;
      x1 = (x1 > 0.f) ? x1 : 0.2f * x1;
      float w0 = a0 * __expf(x0);
      float w1 = a1 * __expf(x1);
      s += w0 + w1;
      v2f a; a.x = w0; a.y = w1;
#pragma unroll
      for (int nt = 0; nt < 4; ++nt) {
        v2f b;
        b.x = sfB[(cb)     * SF_PITCH + nt * 16 + Mloc];
        b.y = sfB[(cb + 1) * SF_PITCH + nt * 16 + Mloc];
        c[nt] = __builtin_amdgcn_wmma_f32_16x16x4_f32(false, a, false, b,
                                                      (short)0, c[nt], false, false);
      }
    }
  }

  // softmax denominator: lane holds partial for row (lane&15); fold halves
  s += __shfl_xor(s, 16, 32);
#pragma unroll
  for (int r = 0; r < 8; ++r) {
    float rs  = __shfl(s, r + 8 * hl, 32);
    float inv = 1.0f / rs;
    int grow  = rowBlock + wv * 16 + r + 8 * hl;
#pragma unroll
    for (int nt = 0; nt < 4; ++nt) {
      int col = nt * 16 + Mloc;
      float v = c[nt][r] * inv + bz[col];
      out[(size_t)grow * GAT_H + col] = (v > 0.f) ? v : (__expf(v) - 1.0f);
    }
  }
}

// -------------------- launcher --------------------
extern "C" void kernel_launch(void* const* d_in, const int* in_sizes, int n_in,
                              void* d_out, int out_size, void* d_ws, size_t ws_size,
                              hipStream_t stream) {
  const float* seq = (const float*)d_in[0];
  const float* adj = (const float*)d_in[1];
  // d_in[2] = bias_mat: unused (derivable from adj; halves HBM traffic)
  const float* W   = (const float*)d_in[3];
  const float* a1w = (const float*)d_in[4];
  const float* a1b = (const float*)d_in[5];
  const float* a2w = (const float*)d_in[6];
  const float* a2b = (const float*)d_in[7];
  const float* bz  = (const float*)d_in[8];
  float* out = (float*)d_out;

  float* sf = (float*)d_ws;                  // 8192*64 f32 = 2 MB
  float* f1 = sf + (size_t)GAT_N * GAT_H;    // 8192 f32
  float* f2 = f1 + GAT_N;                    // 8192 f32

  gat_proj<<<GAT_N / K1_ROWS, 64, 0, stream>>>(seq, W, sf);
  gat_f1f2<<<GAT_N / 8, 256, 0, stream>>>(sf, a1w, a1b, a2w, a2b, f1, f2);
  gat_agg<<<GAT_N / K2_ROWS, 128, (size_t)(3 * BUF_F) * sizeof(float), stream>>>(
      adj, sf, f1, f2, bz, out);
}